// GraphConvolution_60559038874088
// MI455X (gfx1250) — compile-verified
//
#include <hip/hip_runtime.h>

#define N_NODES 100000
#define N_EDGES 1600000
#define D 256            // D_IN == D_OUT == 256
#define LDS_STRIDE 260   // 256 + 4 pad -> conflict-free ds_load_b64

typedef __attribute__((ext_vector_type(2))) float v2f;
typedef __attribute__((ext_vector_type(8))) float v8f;

#define WMMA_F32(A, B, C) \
    __builtin_amdgcn_wmma_f32_16x16x4_f32(false, (A), false, (B), (short)0, (C), false, false)

// ---------------------------------------------------------------------------
// Phase 0: zero the aggregation buffer (d_out doubles as agg; fully
// overwritten by the in-place GEMM afterwards).
// ---------------------------------------------------------------------------
__global__ void zero_kernel(float4* __restrict__ p, long n4) {
    long i = (long)blockIdx.x * blockDim.x + threadIdx.x;
    long stride = (long)gridDim.x * blockDim.x;
    float4 z = make_float4(0.f, 0.f, 0.f, 0.f);
    for (; i < n4; i += stride) p[i] = z;
}

// ---------------------------------------------------------------------------
// Phase 1: edge scatter. One 32-lane group per edge; each lane handles 8
// contiguous channels (two float4 loads, 8 global_atomic_add_f32).
// x (102 MB) fits in the 192 MB L2, so gathers are mostly L2 hits.
// ---------------------------------------------------------------------------
__global__ __launch_bounds__(256) void scatter_kernel(
        const float* __restrict__ x,
        const int*   __restrict__ esrc,
        const int*   __restrict__ edst,
        const float* __restrict__ eval,
        float*       __restrict__ agg) {
    int gid  = blockIdx.x * blockDim.x + threadIdx.x;
    int e    = gid >> 5;            // 32 lanes per edge
    int lane = threadIdx.x & 31;
    if (e >= N_EDGES) return;

    int   s = esrc[e];
    int   d = edst[e];
    float v = eval[e];

    const float4* xs = (const float4*)(x + (long)s * D) + lane * 2;
    float*        ag = agg + (long)d * D + lane * 8;

    float4 a = xs[0];
    float4 b = xs[1];
    atomicAdd(ag + 0, a.x * v);
    atomicAdd(ag + 1, a.y * v);
    atomicAdd(ag + 2, a.z * v);
    atomicAdd(ag + 3, a.w * v);
    atomicAdd(ag + 4, b.x * v);
    atomicAdd(ag + 5, b.y * v);
    atomicAdd(ag + 6, b.z * v);
    atomicAdd(ag + 7, b.w * v);
}

// ---------------------------------------------------------------------------
// Phase 2: in-place GEMM  io[32 rows] = io[32 rows] @ W  per block.
// Block = 128 threads = 4 waves. Wave w owns N-range [w*64, w*64+64) and
// both 16-row M-tiles: 8 accumulators of V_WMMA_F32_16X16X4_F32.
// The 8 scalar B loads per K-step now feed 8 WMMAs (2x density vs 16-row
// tiles). A tile staged to LDS first so the in-place overwrite is safe and
// all 4 waves share the A fragments via LDS broadcast.
// ---------------------------------------------------------------------------
__global__ __launch_bounds__(128) void gemm_kernel(
        float* __restrict__ io, const float* __restrict__ w) {
    __shared__ float lds[32 * LDS_STRIDE];

    int tid  = threadIdx.x;
    int wave = tid >> 5;
    int lane = tid & 31;
    int lg   = lane >> 4;      // lane group: 0 -> K+0/1, 1 -> K+2/3
    int lm   = lane & 15;
    long m0  = (long)blockIdx.x * 32;

    // Stage A (32 x 256 f32) into padded LDS; coalesced float4 copies.
    const float4* asrc = (const float4*)(io + m0 * D);
    for (int v = tid; v < 32 * 64; v += 128) {
        int row = v >> 6;
        int c4  = v & 63;
        float4 t = asrc[v];
        float* drow = lds + row * LDS_STRIDE + c4 * 4;
        drow[0] = t.x; drow[1] = t.y; drow[2] = t.z; drow[3] = t.w;
    }
    __syncthreads();

    int n0 = wave * 64;
    v8f c00 = {}, c01 = {}, c02 = {}, c03 = {};   // M-tile 0, N-tiles 0..3
    v8f c10 = {}, c11 = {}, c12 = {}, c13 = {};   // M-tile 1, N-tiles 0..3

    const float* arow0 = lds + lm * LDS_STRIDE + 2 * lg;       // rows 0..15
    const float* arow1 = arow0 + 16 * LDS_STRIDE;              // rows 16..31
    const float* wbase = w + n0 + lm;                          // B column base

    #pragma unroll 4
    for (int k = 0; k < D; k += 4) {
        // A fragments: lane (lg,lm) holds A[m][k+2lg], A[m][k+2lg+1]
        v2f a0 = *(const v2f*)(arow0 + k);   // 8B-aligned LDS, conflict-free
        v2f a1 = *(const v2f*)(arow1 + k);

        // B fragments: VGPR0 = W[k+2lg][n], VGPR1 = W[k+2lg+1][n]
        int r = k + 2 * lg;
        const float* wr0 = wbase + (long)r * D;
        const float* wr1 = wr0 + D;
        v2f b0 = { wr0[0],  wr1[0]  };
        v2f b1 = { wr0[16], wr1[16] };
        v2f b2 = { wr0[32], wr1[32] };
        v2f b3 = { wr0[48], wr1[48] };

        c00 = WMMA_F32(a0, b0, c00);
        c01 = WMMA_F32(a0, b1, c01);
        c02 = WMMA_F32(a0, b2, c02);
        c03 = WMMA_F32(a0, b3, c03);
        c10 = WMMA_F32(a1, b0, c10);
        c11 = WMMA_F32(a1, b1, c11);
        c12 = WMMA_F32(a1, b2, c12);
        c13 = WMMA_F32(a1, b3, c13);
    }

    // C/D layout: VGPR j -> row (mtile*16 + j + 8*lg), column n0 + 16*t + lm.
    float* orow = io + m0 * D + n0 + lm;
    #pragma unroll
    for (int j = 0; j < 8; ++j) {
        long row0 = (long)(j + 8 * lg) * D;
        orow[row0 + 0]  = c00[j];
        orow[row0 + 16] = c01[j];
        orow[row0 + 32] = c02[j];
        orow[row0 + 48] = c03[j];
        long row1 = (long)(16 + j + 8 * lg) * D;
        orow[row1 + 0]  = c10[j];
        orow[row1 + 16] = c11[j];
        orow[row1 + 32] = c12[j];
        orow[row1 + 48] = c13[j];
    }
}

// ---------------------------------------------------------------------------
extern "C" void kernel_launch(void* const* d_in, const int* in_sizes, int n_in,
                              void* d_out, int out_size, void* d_ws, size_t ws_size,
                              hipStream_t stream) {
    const float* x    = (const float*)d_in[0];
    const int*   esrc = (const int*)  d_in[1];
    const int*   edst = (const int*)  d_in[2];
    const float* ev   = (const float*)d_in[3];
    const float* w    = (const float*)d_in[4];
    float*       out  = (float*)d_out;   // serves as agg, then overwritten in place

    (void)in_sizes; (void)n_in; (void)out_size; (void)d_ws; (void)ws_size;

    long n4 = (long)N_NODES * D / 4;
    zero_kernel<<<2048, 256, 0, stream>>>((float4*)out, n4);

    int sblocks = (N_EDGES + 7) / 8;   // 8 edges per 256-thread block
    scatter_kernel<<<sblocks, 256, 0, stream>>>(x, esrc, edst, ev, out);

    gemm_kernel<<<N_NODES / 32, 128, 0, stream>>>(out, w);
}